// Zigzag_iteration_80685255623029
// MI455X (gfx1250) — compile-verified
//
#include <hip/hip_runtime.h>
#include <math.h>

typedef float v2f __attribute__((ext_vector_type(2)));
typedef float v8f __attribute__((ext_vector_type(8)));

#define NEGV -1e9f

#if __has_builtin(__builtin_amdgcn_tensor_load_to_lds) && \
    __has_builtin(__builtin_amdgcn_s_wait_tensorcnt)
#define HAVE_TDM 1
typedef unsigned int v4u __attribute__((ext_vector_type(4)));
typedef int v4i __attribute__((ext_vector_type(4)));
typedef int v8i __attribute__((ext_vector_type(8)));

// Issue one TDM 2D tile load: 64 rows x 16 dwords, +1 dword LDS pad per row
// (matches float[64][17] LDS tile). Rows beyond rowsAvail zero-fill via the
// descriptor's tensor_dim1 OOB rule; kAvail >= 16 always in our call sites.
// 6-arg builtin: (uint32x4 g0, int32x8 g1, int32x4 g2, int32x4 g3, int32x8 g4, i32 cpol)
__device__ __attribute__((always_inline))
static void tdm_load_tile(unsigned ldsAddr, const float* gptr,
                          unsigned kAvail, unsigned rowsAvail,
                          unsigned long long rowStrideElems)
{
    unsigned long long ga = (unsigned long long)gptr;
    v4u g0;
    g0.x = 1u;                                            // count=1, user desc
    g0.y = ldsAddr;                                       // lds_addr (bytes)
    g0.z = (unsigned)ga;                                  // global_addr[31:0]
    g0.w = (unsigned)((ga >> 32) & 0x1FFFFFFull) | (2u << 30); // addr[56:32], type=2
    v8i g1;
    g1[0] = (int)((2u << 16)    // data_size = 4 bytes
                | (1u << 20)    // pad_enable
                | (3u << 22));  // pad_interval: 16 dwords (pad_amount=0 -> 1 dword)
    g1[1] = (int)((kAvail & 0xFFFFu) << 16);                       // tensor_dim0 lo16
    g1[2] = (int)((kAvail >> 16) | ((rowsAvail & 0xFFFFu) << 16)); // td0 hi | td1 lo
    g1[3] = (int)((rowsAvail >> 16) | (16u << 16));                // td1 hi | tile_dim0=16
    g1[4] = 64;                                                    // tile_dim1=64, tile_dim2=0
    g1[5] = (int)(unsigned)rowStrideElems;                         // dim0_stride[31:0]
    g1[6] = (int)((rowStrideElems >> 32) & 0xFFFFull);             // dim0_stride[47:32]
    g1[7] = 0;
    v4i z4 = {0, 0, 0, 0};
    v8i z8 = {0, 0, 0, 0, 0, 0, 0, 0};
    __builtin_amdgcn_tensor_load_to_lds(g0, g1, z4, z4, z8, 0);
}
#endif

// 16x16x4 fp32 WMMA over one staged 64x16 k-chunk.
__device__ __attribute__((always_inline))
static void mma_tile(const float As[64][17], const float Bs[64][17],
                     v8f acc[2][2], int wm, int wn, int lm, int lk)
{
#pragma unroll
    for (int kk = 0; kk < 16; kk += 4) {
        v2f a0, a1, b0, b1;
        a0.x = As[wm + lm][kk + lk];      a0.y = As[wm + lm][kk + lk + 1];
        a1.x = As[wm + 16 + lm][kk + lk]; a1.y = As[wm + 16 + lm][kk + lk + 1];
        b0.x = Bs[wn + lm][kk + lk];      b0.y = Bs[wn + lm][kk + lk + 1];
        b1.x = Bs[wn + 16 + lm][kk + lk]; b1.y = Bs[wn + 16 + lm][kk + lk + 1];
        acc[0][0] = __builtin_amdgcn_wmma_f32_16x16x4_f32(false, a0, false, b0, (short)0, acc[0][0], false, false);
        acc[0][1] = __builtin_amdgcn_wmma_f32_16x16x4_f32(false, a0, false, b1, (short)0, acc[0][1], false, false);
        acc[1][0] = __builtin_amdgcn_wmma_f32_16x16x4_f32(false, a1, false, b0, (short)0, acc[1][0], false, false);
        acc[1][1] = __builtin_amdgcn_wmma_f32_16x16x4_f32(false, a1, false, b1, (short)0, acc[1][1], false, false);
    }
}

// =====================================================================
// Generic strided, batched fp32 WMMA GEMM:
//   C[z,m,n] = epilogue( sum_k A[z,m,k] * B[z,n,k] )
// A addressed as aoff + m*ars + k*aks ; B as boff + n*brs + k*bks
// C column stride is 1, row stride crs.
// act: 0=none(+bias), 1=relu(+bias), 2=sigmoid(+bias)*gate[m,n]
// When aks==bks==1, tiles are staged via a double-buffered Tensor Data
// Mover pipeline (DMA of chunk i+1 overlaps WMMA on chunk i; TDM ops
// complete in order per wave so s_wait_tensorcnt 2 == chunk i landed).
// =====================================================================
__global__ __launch_bounds__(128)
void gemm_f32_wmma(const float* __restrict__ A,
                   const float* __restrict__ Bm,
                   float* __restrict__ C,
                   int M, int N, int K,
                   long long ab, long long bb, long long cb,
                   long long ars, long long aks,
                   long long brs, long long bks,
                   long long crs,
                   const float* __restrict__ bias,
                   const float* __restrict__ gate,
                   long long gb, int act)
{
    __shared__ float As[2][64][17];
    __shared__ float Bs[2][64][17];

    const int tid  = threadIdx.x;
    const int lane = tid & 31;
    const int wave = tid >> 5;
    const int wm   = (wave >> 1) << 5;   // 0 or 32
    const int wn   = (wave & 1) << 5;    // 0 or 32
    const int lm   = lane & 15;
    const int lk   = (lane >> 4) << 1;   // 0 or 2
    const int lr8  = (lane >> 4) << 3;   // 0 or 8

    const int tN = blockIdx.x * 64;
    const int tM = blockIdx.y * 64;
    const long long zb   = blockIdx.z;
    const long long aoff = zb * ab, boff = zb * bb, coff = zb * cb;

    v8f acc[2][2];
#pragma unroll
    for (int i = 0; i < 2; ++i)
#pragma unroll
        for (int j = 0; j < 2; ++j)
#pragma unroll
            for (int r = 0; r < 8; ++r) acc[i][j][r] = 0.f;

#ifdef HAVE_TDM
    if (aks == 1 && bks == 1) {
        // ---------- double-buffered TDM pipeline (unit k-stride) ----------
        const unsigned ldsA[2] = { (unsigned)(unsigned long long)(&As[0][0][0]),
                                   (unsigned)(unsigned long long)(&As[1][0][0]) };
        const unsigned ldsB[2] = { (unsigned)(unsigned long long)(&Bs[0][0][0]),
                                   (unsigned)(unsigned long long)(&Bs[1][0][0]) };
        const float* aBase = A + aoff + (long long)tM * ars;
        const float* bBase = Bm + boff + (long long)tN * brs;
        const unsigned aRows = (unsigned)(M - tM);   // >0 by grid construction
        const unsigned bRows = (unsigned)(N - tN);
        const bool issuer = (wave == 0);   // correct whether branched or exec-masked
        const int nChunks = (K + 15) >> 4;

        if (issuer) {
            tdm_load_tile(ldsA[0], aBase, (unsigned)K, aRows, (unsigned long long)ars);
            tdm_load_tile(ldsB[0], bBase, (unsigned)K, bRows, (unsigned long long)brs);
        }
        for (int ci = 0; ci < nChunks; ++ci) {
            const int buf = ci & 1;
            if (ci + 1 < nChunks) {
                const int k1 = (ci + 1) << 4;
                if (issuer) {
                    tdm_load_tile(ldsA[buf ^ 1], aBase + k1, (unsigned)(K - k1),
                                  aRows, (unsigned long long)ars);
                    tdm_load_tile(ldsB[buf ^ 1], bBase + k1, (unsigned)(K - k1),
                                  bRows, (unsigned long long)brs);
                }
                __builtin_amdgcn_s_wait_tensorcnt(2);   // chunk ci's 2 loads done
            } else {
                __builtin_amdgcn_s_wait_tensorcnt(0);
            }
            __syncthreads();
            mma_tile(As[buf], Bs[buf], acc, wm, wn, lm, lk);
            __syncthreads();
        }
    } else
#endif
    {
        // ---------- Generic strided fallback ----------
        for (int k0 = 0; k0 < K; k0 += 16) {
#pragma unroll
            for (int i = 0; i < 8; ++i) {
                int idx = tid + (i << 7);
                int r = idx >> 4, c = idx & 15;
                int gk = k0 + c;
                int gm = tM + r;
                As[0][r][c] = (gm < M && gk < K)
                    ? A[aoff + (long long)gm * ars + (long long)gk * aks] : 0.f;
                int gn = tN + r;
                Bs[0][r][c] = (gn < N && gk < K)
                    ? Bm[boff + (long long)gn * brs + (long long)gk * bks] : 0.f;
            }
            __syncthreads();
            mma_tile(As[0], Bs[0], acc, wm, wn, lm, lk);
            __syncthreads();
        }
    }

#pragma unroll
    for (int fm = 0; fm < 2; ++fm)
#pragma unroll
        for (int fn = 0; fn < 2; ++fn)
#pragma unroll
            for (int r = 0; r < 8; ++r) {
                int row = tM + wm + (fm << 4) + r + lr8;
                int col = tN + wn + (fn << 4) + lm;
                if (row < M && col < N) {
                    float v = acc[fm][fn][r];
                    if (bias) v += bias[col];
                    if (act == 1) {
                        v = fmaxf(v, 0.f);
                    } else if (act == 2) {
                        float g = gate[zb * gb + (long long)row * crs + col];
                        v = g / (1.f + expf(-v));
                    }
                    C[coff + (long long)row * crs + col] = v;
                }
            }
}

// =====================================================================
// Elementwise helpers
// =====================================================================
__global__ void bmat_kernel(const float* __restrict__ Eq,
                            const float* __restrict__ W_in,
                            float* __restrict__ Bm, int n)
{
    int idx = blockIdx.x * 256 + threadIdx.x;
    if (idx >= n) return;
    int h  = idx & 255;
    int t  = (idx >> 8) & 31;
    int bq = idx >> 13;                 // b*192+q
    Bm[idx] = Eq[bq * 256 + h] * W_in[t * 768 + 512 + h];
}

__global__ void e1_add_kernel(float* __restrict__ E1,
                              const float* __restrict__ Aq,
                              const float* __restrict__ Ap, int n)
{
    int idx = blockIdx.x * 256 + threadIdx.x;
    if (idx >= n) return;
    int t   = idx & 31;
    int tmp = idx >> 5;                 // (b*192+p)*192 + q
    int q   = tmp % 192;
    int bp  = tmp / 192;                // b*192+p
    int b   = bp / 192;
    E1[idx] += Aq[(b * 192 + q) * 32 + t] + Ap[bp * 32 + t];
}

__global__ void obv_kernel(const float* __restrict__ hid,
                           const float* __restrict__ Wo2,
                           const float* __restrict__ bo2,
                           float* __restrict__ obv, int n)
{
    int idx = blockIdx.x * 256 + threadIdx.x;
    if (idx >= n) return;
    const float* h = hid + (long long)idx * 32;
    float s = bo2[0];
#pragma unroll
    for (int t = 0; t < 32; ++t) s += h[t] * Wo2[t];
    obv[idx] = 1.f / (1.f + expf(-s));
}

__global__ void eqw_kernel(const float* __restrict__ Eq,
                           const float* __restrict__ W_out,
                           float* __restrict__ Eqw, int n)
{
    int idx = blockIdx.x * 256 + threadIdx.x;
    if (idx >= n) return;
    int h = idx & 255;
    Eqw[idx] = Eq[idx] * W_out[512 + h];
}

__global__ void uqup_kernel(const float* __restrict__ Eq,
                            const float* __restrict__ Ep,
                            const float* __restrict__ W_out,
                            float* __restrict__ uq,
                            float* __restrict__ up, int n)
{
    int idx = blockIdx.x * 256 + threadIdx.x;
    if (idx >= n) return;
    if (idx < 768) {
        const float* e = Eq + (long long)idx * 256;
        float s = 0.f;
        for (int h = 0; h < 256; ++h) s += e[h] * W_out[h];
        uq[idx] = s;
    } else {
        int r = idx - 768;
        const float* e = Ep + (long long)r * 256;
        float s = 0.f;
        for (int h = 0; h < 256; ++h) s += e[h] * W_out[256 + h];
        up[r] = s;
    }
}

__global__ void u_kernel(const float* __restrict__ uq,
                         const float* __restrict__ up,
                         const float* __restrict__ cross,
                         const float* __restrict__ Mb,
                         const float* __restrict__ W_out,
                         const float* __restrict__ mask,
                         float* __restrict__ U, int n)
{
    int idx = blockIdx.x * 256 + threadIdx.x;
    if (idx >= n) return;
    int b = idx / 36864;
    int w = idx % 36864;
    int p = w / 192, q = w % 192;
    float u = uq[b * 192 + q] + up[b * 192 + p] + cross[idx];
    const float* m = Mb + (long long)idx * 32;
#pragma unroll
    for (int t = 0; t < 32; ++t) u += m[t] * W_out[768 + t];
    if (mask[idx] < 0.5f) u = NEGV;
    U[idx] = u;
}

__global__ void gbuild_kernel(const float* __restrict__ Eb,
                              const float* __restrict__ Xb,
                              float* __restrict__ G, int n)
{
    int idx = blockIdx.x * 256 + threadIdx.x;
    if (idx >= n) return;
    int r = idx >> 8;
    int h = idx & 255;
    float e = Eb[idx], x = Xb[idx];
    long long base = (long long)r * 768 + h;
    G[base]       = e;
    G[base + 256] = x;
    G[base + 512] = e * x;
}

// =====================================================================
// Axis attention (192 keys, head dim 32), online softmax per query.
//  logits = (q.k)/sqrt(32) * obv[k] ; mask<0.5 -> -1e9 ; out = softmax @ v
// block = (b,i) row; thread j = one query.
// =====================================================================
__global__ __launch_bounds__(192)
void axis_attention(const float* __restrict__ fq,
                    const float* __restrict__ fk,
                    const float* __restrict__ fv,
                    const float* __restrict__ obv,
                    const float* __restrict__ mask,
                    float* __restrict__ Mout,
                    int vertical, int accumulate)
{
    __shared__ float sK[192 * 32];
    __shared__ float sV[192 * 32];
    __shared__ float sO[192];
    __shared__ float sM[192];

    int i = blockIdx.x, b = blockIdx.y, j = threadIdx.x;
    long long qbase; long long sj; long long obase; long long os;
    if (!vertical) {
        qbase = (long long)(b * 192 + i) * 192 * 32; sj = 32;
        obase = (long long)(b * 192 + i) * 192;      os = 1;
    } else {
        qbase = (long long)b * 1179648 + (long long)i * 32; sj = 6144;
        obase = (long long)b * 36864 + i;                    os = 192;
    }

#pragma unroll
    for (int t = 0; t < 32; ++t) {
        sK[j * 32 + t] = fk[qbase + (long long)j * sj + t];
        sV[j * 32 + t] = fv[qbase + (long long)j * sj + t];
    }
    sO[j] = obv[obase + (long long)j * os];
    sM[j] = mask[obase + (long long)j * os];
    __syncthreads();

    float q[32];
#pragma unroll
    for (int t = 0; t < 32; ++t) q[t] = fq[qbase + (long long)j * sj + t];

    const float scale = 0.17677669529663687f;   // 1/sqrt(32)
    float mrun = -INFINITY, s = 0.f;
    float acc[32];
#pragma unroll
    for (int t = 0; t < 32; ++t) acc[t] = 0.f;

    for (int kk = 0; kk < 192; ++kk) {
        float d = 0.f;
#pragma unroll
        for (int t = 0; t < 32; ++t) d += q[t] * sK[kk * 32 + t];
        float l = d * scale * sO[kk];
        if (sM[kk] < 0.5f) l = NEGV;
        float nm = fmaxf(mrun, l);
        float f  = expf(mrun - nm);
        float p  = expf(l - nm);
        s = s * f + p;
#pragma unroll
        for (int t = 0; t < 32; ++t) acc[t] = acc[t] * f + p * sV[kk * 32 + t];
        mrun = nm;
    }
    float inv = 1.f / s;
#pragma unroll
    for (int t = 0; t < 32; ++t) {
        long long oi = qbase + (long long)j * sj + t;
        float o = acc[t] * inv;
        Mout[oi] = accumulate ? (Mout[oi] + o) : o;
    }
}

// =====================================================================
// Row/column softmax over length-192 vectors.
// colmode=0: rows (b,p) over q stride 1 ; colmode=1: cols (b,q) over p stride 192
// =====================================================================
__global__ __launch_bounds__(256)
void softmax_axis(const float* __restrict__ U, float* __restrict__ out, int colmode)
{
    __shared__ float sh[256];
    int r = blockIdx.x;                 // 0..767
    int b = r / 192, i = r % 192;
    long long base; long long stride;
    if (!colmode) { base = (long long)r * 192;          stride = 1;   }
    else          { base = (long long)b * 36864 + i;    stride = 192; }
    int t = threadIdx.x;
    float v = (t < 192) ? U[base + (long long)t * stride] : -INFINITY;
    sh[t] = v;
    __syncthreads();
    for (int o = 128; o > 0; o >>= 1) {
        if (t < o) sh[t] = fmaxf(sh[t], sh[t + o]);
        __syncthreads();
    }
    float mx = sh[0];
    __syncthreads();
    float e = (t < 192) ? expf(v - mx) : 0.f;
    sh[t] = e;
    __syncthreads();
    for (int o = 128; o > 0; o >>= 1) {
        if (t < o) sh[t] += sh[t + o];
        __syncthreads();
    }
    float inv = 1.f / sh[0];
    if (t < 192) out[base + (long long)t * stride] = e * inv;
}

// =====================================================================
// Host launcher
// =====================================================================
static inline void launch_gemm(hipStream_t s,
                               const float* A, const float* B, float* C,
                               int M, int N, int K, int batches,
                               long long ab, long long bb, long long cb,
                               long long ars, long long aks,
                               long long brs, long long bks, long long crs,
                               const float* bias, const float* gate,
                               long long gb, int act)
{
    dim3 grid((N + 63) / 64, (M + 63) / 64, batches);
    gemm_f32_wmma<<<grid, dim3(128), 0, s>>>(A, B, C, M, N, K, ab, bb, cb,
                                             ars, aks, brs, bks, crs,
                                             bias, gate, gb, act);
}

extern "C" void kernel_launch(void* const* d_in, const int* in_sizes, int n_in,
                              void* d_out, int out_size, void* d_ws, size_t ws_size,
                              hipStream_t stream)
{
    (void)in_sizes; (void)n_in; (void)out_size; (void)ws_size;
    const float* Eq    = (const float*)d_in[0];   // [4,192,256]
    const float* Ep    = (const float*)d_in[1];   // [4,192,256]
    const float* mask  = (const float*)d_in[2];   // [4,192,192]
    const float* W_in  = (const float*)d_in[3];   // [32,768]
    const float* Wq    = (const float*)d_in[4];
    const float* bq    = (const float*)d_in[5];
    const float* Wk    = (const float*)d_in[6];
    const float* bk    = (const float*)d_in[7];
    const float* Wv    = (const float*)d_in[8];
    const float* bv    = (const float*)d_in[9];
    const float* Wo1   = (const float*)d_in[10];
    const float* bo1   = (const float*)d_in[11];
    const float* Wo2   = (const float*)d_in[12];  // [1,32]
    const float* bo2   = (const float*)d_in[13];  // [1]
    const float* W_out = (const float*)d_in[14];  // [1,800]
    const float* Wl    = (const float*)d_in[15];  // [768,768]
    const float* bl    = (const float*)d_in[16];
    const float* Wr    = (const float*)d_in[17];
    const float* br    = (const float*)d_in[18];
    float* out = (float*)d_out;                   // [4,192,192]

    float* ws  = (float*)d_ws;
    float* Aq  = ws;                       // 24576
    float* Ap  = Aq + 24576;               // 24576
    float* Bm  = Ap + 24576;               // 6291456  (later aliased by hid)
    float* hid = Bm;                       // 4718592 alias
    float* E1  = Bm + 6291456;             // 4718592  (later aliased by M)
    float* Mb  = E1;                       // alias
    float* fq  = E1 + 4718592;             // 4718592  (post-attn pool)
    float* fk  = fq + 4718592;
    float* fv  = fk + 4718592;
    float* obv = fv + 4718592;             // 147456
    // post-attention reuse pool inside fq region:
    float* Eqw   = fq;                     // 196608
    float* cross = Eqw + 196608;           // 147456
    float* uq    = cross + 147456;         // 768
    float* up    = uq + 768;               // 768
    float* Ub    = up + 768;               // 147456
    float* A_p   = Ub + 147456;            // 147456
    float* B_p   = A_p + 147456;           // 147456
    float* App   = B_p + 147456;           // 196608
    float* Bpp   = App + 196608;           // 196608
    float* Gq    = Bpp + 196608;           // 589824
    float* Gp    = Gq + 589824;            // 589824
    float* lft   = Gp + 589824;            // 589824
    float* rgt   = lft + 589824;           // 589824

    // 1) Aq = Eq @ W1^T, Ap = Ep @ W2^T   (M=768,N=32,K=256)
    launch_gemm(stream, Eq, W_in,       Aq, 768, 32, 256, 1, 0, 0, 0,
                256, 1, 768, 1, 32, nullptr, nullptr, 0, 0);
    launch_gemm(stream, Ep, W_in + 256, Ap, 768, 32, 256, 1, 0, 0, 0,
                256, 1, 768, 1, 32, nullptr, nullptr, 0, 0);

    // 2) Bmat[(b,q,t),h] = Eq * W3
    bmat_kernel<<<(6291456 + 255) / 256, 256, 0, stream>>>(Eq, W_in, Bm, 6291456);

    // 3) E1_prod[b] = Ep[b] (192x256) @ Bmat[b]^T -> [192,6144]
    launch_gemm(stream, Ep, Bm, E1, 192, 6144, 256, 4,
                192LL * 256, 6144LL * 256, 192LL * 6144,
                256, 1, 256, 1, 6144, nullptr, nullptr, 0, 0);

    // 4) E1 += Aq + Ap
    e1_add_kernel<<<(4718592 + 255) / 256, 256, 0, stream>>>(E1, Aq, Ap, 4718592);

    // 5) fq/fk/fv/hid : [147456,32] @ [32,32]^T
    launch_gemm(stream, E1, Wq,  fq,  147456, 32, 32, 1, 0, 0, 0,
                32, 1, 32, 1, 32, bq, nullptr, 0, 0);
    launch_gemm(stream, E1, Wk,  fk,  147456, 32, 32, 1, 0, 0, 0,
                32, 1, 32, 1, 32, bk, nullptr, 0, 0);
    launch_gemm(stream, E1, Wv,  fv,  147456, 32, 32, 1, 0, 0, 0,
                32, 1, 32, 1, 32, bv, nullptr, 0, 0);
    launch_gemm(stream, E1, Wo1, hid, 147456, 32, 32, 1, 0, 0, 0,
                32, 1, 32, 1, 32, bo1, nullptr, 0, 1);   // relu

    // 6) obv = sigmoid(hid @ Wo2 + bo2)
    obv_kernel<<<(147456 + 255) / 256, 256, 0, stream>>>(hid, Wo2, bo2, obv, 147456);

    // 7/8) M = horizontal + vertical attention (M aliases E1)
    dim3 ag(192, 4);
    axis_attention<<<ag, 192, 0, stream>>>(fq, fk, fv, obv, mask, Mb, 0, 0);
    axis_attention<<<ag, 192, 0, stream>>>(fq, fk, fv, obv, mask, Mb, 1, 1);

    // 9) Eqw = Eq * w3 ; cross[b] = Ep[b] @ Eqw[b]^T
    eqw_kernel<<<(196608 + 255) / 256, 256, 0, stream>>>(Eq, W_out, Eqw, 196608);
    launch_gemm(stream, Ep, Eqw, cross, 192, 192, 256, 4,
                192LL * 256, 192LL * 256, 192LL * 192,
                256, 1, 256, 1, 192, nullptr, nullptr, 0, 0);

    // 10) uq, up ; 11) U
    uqup_kernel<<<(1536 + 255) / 256, 256, 0, stream>>>(Eq, Ep, W_out, uq, up, 1536);
    u_kernel<<<(147456 + 255) / 256, 256, 0, stream>>>(uq, up, cross, Mb, W_out,
                                                       mask, Ub, 147456);

    // 12) softmaxes
    softmax_axis<<<768, 256, 0, stream>>>(Ub, A_p, 0);   // over q
    softmax_axis<<<768, 256, 0, stream>>>(Ub, B_p, 1);   // over p

    // 13) A__p[b] = A_p[b] @ Eq[b] ; B__p[b] = B_p[b]^T @ Ep[b]  (strided fallback path)
    launch_gemm(stream, A_p, Eq, App, 192, 256, 192, 4,
                192LL * 192, 192LL * 256, 192LL * 256,
                192, 1, 1, 256, 256, nullptr, nullptr, 0, 0);
    launch_gemm(stream, B_p, Ep, Bpp, 192, 256, 192, 4,
                192LL * 192, 192LL * 256, 192LL * 256,
                1, 192, 1, 256, 256, nullptr, nullptr, 0, 0);

    // 14) gate inputs
    gbuild_kernel<<<(196608 + 255) / 256, 256, 0, stream>>>(Ep, App, Gq, 196608);
    gbuild_kernel<<<(196608 + 255) / 256, 256, 0, stream>>>(Eq, Bpp, Gp, 196608);

    // 15) left/right = sigmoid(G @ W^T + b) * G   (M=768,N=768,K=768)
    launch_gemm(stream, Gq, Wl, lft, 768, 768, 768, 1, 0, 0, 0,
                768, 1, 768, 1, 768, bl, Gq, 0, 2);
    launch_gemm(stream, Gp, Wr, rgt, 768, 768, 768, 1, 0, 0, 0,
                768, 1, 768, 1, 768, br, Gp, 0, 2);

    // 16) out[b] = relu(left[b] @ right[b]^T)
    launch_gemm(stream, lft, rgt, out, 192, 192, 768, 4,
                192LL * 768, 192LL * 768, 192LL * 192,
                768, 1, 768, 1, 192, nullptr, nullptr, 0, 1);
}